// NonSharedConv2d_10737418240811
// MI455X (gfx1250) — compile-verified
//
#include <hip/hip_runtime.h>

// NonSharedConv2d (locally-connected conv) on gfx1250.
// B=4, IC=64, H=W=32, OC=64, K=3, OH=OW=30, FAN_IN=576.
// Memory-bound: 531 MB of once-used weights -> stream NT at HBM rate.
// Math kept in fp32 via V_WMMA_F32_16X16X4_F32 (exact vs reference).

typedef float v2f __attribute__((ext_vector_type(2)));
typedef float v8f __attribute__((ext_vector_type(8)));

#define BATCH 4
#define IC    64
#define IH    32
#define IW    32
#define OC    64
#define OHH   30
#define OWW   30
#define FAN   576   // IC * 3 * 3

__global__ __launch_bounds__(128)
void nsconv_wmma_kernel(const float* __restrict__ x,
                        const float* __restrict__ wts,
                        float* __restrict__ out)
{
    __shared__ float patch[FAN];

    const int pos = blockIdx.x;                 // 0..3599 = (b, oi, oj)
    const int b   = pos / (OHH * OWW);
    const int ij  = pos - b * (OHH * OWW);
    const int oi  = ij / OWW;
    const int oj  = ij - oi * OWW;

    // ---- Stage the im2col patch (c, kh, kw order) into LDS ----
    for (int p = threadIdx.x; p < FAN; p += 128) {
        int c  = p / 9;
        int r  = p - c * 9;
        int kh = r / 3;
        int kw = r - kh * 3;
        patch[p] = x[((b * IC + c) * IH + (oi + kh)) * IW + (oj + kw)];
    }
    __syncthreads();

    const int lane = threadIdx.x & 31;
    const int wv   = threadIdx.x >> 5;          // wave 0..3 -> OC tile
    const int row  = lane & 15;                 // M row within 16x4 A tile
    const int kofs = (lane >> 4) << 1;          // lo lanes: K+0/K+1, hi: K+2/K+3

    // weights layout: [pos][oc][fan] ; A row for this lane:
    const float* __restrict__ arow =
        wts + (size_t)pos * (OC * FAN) + (size_t)(wv * 16 + row) * FAN + kofs;

    v8f acc0 = {};
    v8f acc1 = {};

    // 144 K-steps of 16x16x4; two accumulators to break the WMMA RAW chain.
    #pragma unroll 4
    for (int k = 0; k < FAN; k += 8) {
        // A: 8B per lane, strided rows; stream non-temporal (used once).
        v2f a0 = __builtin_nontemporal_load((const v2f*)(arow + k));
        // B: broadcast patch segment from LDS (8B-aligned: k%4==0, kofs even).
        v2f b0 = *(const v2f*)(&patch[k + kofs]);
        acc0 = __builtin_amdgcn_wmma_f32_16x16x4_f32(
                   false, a0, false, b0, (short)0, acc0, false, false);

        v2f a1 = __builtin_nontemporal_load((const v2f*)(arow + k + 4));
        v2f b1 = *(const v2f*)(&patch[k + 4 + kofs]);
        acc1 = __builtin_amdgcn_wmma_f32_16x16x4_f32(
                   false, a1, false, b1, (short)0, acc1, false, false);
    }
    v8f acc = acc0 + acc1;

    // D layout: VGPR v -> M=v (lanes 0-15) / M=v+8 (lanes 16-31); all N equal.
    if (row == 0) {
        const int ocbase = wv * 16 + ((lane >> 4) << 3);
        #pragma unroll
        for (int v = 0; v < 8; ++v) {
            out[((b * OC + (ocbase + v)) * OHH + oi) * OWW + oj] = acc[v];
        }
    }
}

extern "C" void kernel_launch(void* const* d_in, const int* in_sizes, int n_in,
                              void* d_out, int out_size, void* d_ws, size_t ws_size,
                              hipStream_t stream)
{
    const float* x   = (const float*)d_in[0];   // (4, 64, 32, 32)
    const float* wts = (const float*)d_in[1];   // (4, 30, 30, 64, 576)
    float*       out = (float*)d_out;           // (4, 64, 30, 30)

    dim3 grid(BATCH * OHH * OWW);               // 3600 positions
    dim3 block(128);                            // 4 waves: one 16-OC tile each
    nsconv_wmma_kernel<<<grid, block, 0, stream>>>(x, wts, out);
}